// GCN_pool_layers_39513699123566
// MI455X (gfx1250) — compile-verified
//
#include <hip/hip_runtime.h>
#include <hip/hip_bf16.h>

typedef float v2f __attribute__((ext_vector_type(2)));
typedef float v8f __attribute__((ext_vector_type(8)));

#define Bg   512
#define Nn   1024
#define DEGc 16
#define EPG  (Nn*DEGc)       // 16384 edges per graph (block-contiguous)
#define ETOT (Bg*EPG)        // 8388608 total edges
#define FIN  128
#define EMB  32
#define HALF 16
#define STR  20              // padded row stride (floats): banks (20d+c)%64 -> 16 groups,
                             // ~2-way conflicts vs 8-way at stride 16; 80B keeps 16B align
#define KTOP 820
#define BD   256             // 8 wave32

__global__ __launch_bounds__(BD, 1)
void gcn_topk_fused(const float* __restrict__ x,      // [B*N, 128]
                    const int*   __restrict__ e32,    // int64 edge_index viewed as i32 pairs
                    const float* __restrict__ W_in,   // [128,32] row-major
                    const float* __restrict__ b_in,   // [32]
                    const float* __restrict__ pool_w, // [32]
                    const float* __restrict__ W_out,  // [32]
                    const float* __restrict__ b_out,  // [1]
                    float* __restrict__ out)          // [512]
{
  __shared__ float Wh[FIN*HALF];   //  8 KB  W half, [k][c] (stride 16: GEMM reads conflict-free)
  __shared__ float hs[Nn*STR];     // 80 KB  h = x@W half (padded rows)
  __shared__ float acc[Nn*STR];    // 80 KB  scatter accumulator (padded rows)
  __shared__ float dinv[Nn];       //  4 KB  1/sqrt(deg)
  __shared__ float sacc[Nn];       //  4 KB  partial score dot
  __shared__ float zacc[Nn];       //  4 KB  partial out.W_out dot
  __shared__ float skey[Nn];       //  4 KB  sort keys (scores)
  __shared__ float sval[Nn];       //  4 KB  sort values (z)
  __shared__ float red[BD];        //  1 KB  reduction

  const int b    = blockIdx.x;
  const int tid  = threadIdx.x;
  const int lane = tid & 31;
  const int wave = tid >> 5;
  const int lrow = lane & 15;      // M (A) / N (B,C,D) index within 16x16 tile
  const int lhi  = lane >> 4;      // selects K pair group / upper C rows

  const float* xg = x + (size_t)b * Nn * FIN;
  const long long ebase = (long long)b * EPG;

  // ---- degrees (self-loop => init 1) + clear score accumulators ----
  for (int i = tid; i < Nn; i += BD) { dinv[i] = 1.0f; sacc[i] = 0.0f; zacc[i] = 0.0f; }
  __syncthreads();
  for (int e = tid; e < EPG; e += BD) {
    int d = e32[2*((long long)ETOT + ebase + e)] - b*Nn;   // low 32 bits of int64 dst
    atomicAdd(&dinv[d], 1.0f);                              // ds_add_f32
  }
  __syncthreads();
  for (int i = tid; i < Nn; i += BD) dinv[i] = rsqrtf(dinv[i]);

  // ---- two 16-channel passes over EMB ----
  for (int p = 0; p < 2; ++p) {
    __syncthreads();
    for (int i = tid; i < FIN*HALF; i += BD) {              // stage W half to LDS
      int k = i >> 4, c = i & 15;
      Wh[i] = W_in[k*EMB + p*HALF + c];
    }
    __syncthreads();

    // GEMM hs[1024][16] = xg[1024][128] @ Wh[128][16] via V_WMMA_F32_16X16X4_F32.
    // A 16x4 layout: lanes 0-15 = M rows, VGPR pair holds K=(k0+2*lhi, +1).
    // B 4x16 symmetric: lane%16 = N col, same K pair.
    for (int t = wave; t < Nn/16; t += BD/32) {
      v8f c8 = {0.f,0.f,0.f,0.f,0.f,0.f,0.f,0.f};
      const float* xrow = xg + (size_t)(t*16 + lrow) * FIN;
      for (int k0 = 0; k0 < FIN; k0 += 4) {
        int kk = k0 + 2*lhi;
        v2f a;  a.x  = xrow[kk];            a.y  = xrow[kk+1];
        v2f bb; bb.x = Wh[kk*HALF + lrow];  bb.y = Wh[(kk+1)*HALF + lrow];
        c8 = __builtin_amdgcn_wmma_f32_16x16x4_f32(
                 false, a, false, bb, (short)0, c8, false, false);
      }
      #pragma unroll
      for (int v = 0; v < 8; ++v) {         // C/D: vgpr v -> row v + 8*lhi, lane%16 -> col
        int row = t*16 + v + 8*lhi;
        hs[row*STR + lrow] = c8[v];
      }
    }
    __syncthreads();

    // acc init with self-loop term h_i * dinv_i^2
    for (int i = tid; i < Nn; i += BD) {
      float n2 = dinv[i]*dinv[i];
      #pragma unroll
      for (int c = 0; c < HALF; ++c) acc[i*STR + c] = hs[i*STR + c]*n2;
    }
    __syncthreads();

    // edge scatter: acc[dst] += h[src] * dinv[src]*dinv[dst]   (LDS float atomics)
    for (int e = tid; e < EPG; e += BD) {
      int s = e32[2*(ebase + e)]                        - b*Nn;
      int d = e32[2*((long long)ETOT + ebase + e)]      - b*Nn;
      float nm = dinv[s]*dinv[d];
      const float4* hv = reinterpret_cast<const float4*>(&hs[s*STR]); // 16B aligned (80B rows)
      float* ap = &acc[d*STR];
      #pragma unroll
      for (int q = 0; q < 4; ++q) {
        float4 h4 = hv[q];
        atomicAdd(ap + 4*q + 0, h4.x*nm);
        atomicAdd(ap + 4*q + 1, h4.y*nm);
        atomicAdd(ap + 4*q + 2, h4.z*nm);
        atomicAdd(ap + 4*q + 3, h4.w*nm);
      }
    }
    __syncthreads();

    // relu(out) and fold into score / z partial dot products
    for (int i = tid; i < Nn; i += BD) {
      float sp = 0.f, zp = 0.f;
      #pragma unroll
      for (int c = 0; c < HALF; ++c) {
        float o = fmaxf(acc[i*STR + c] + b_in[p*HALF + c], 0.f);
        sp += o * pool_w[p*HALF + c];
        zp += o * W_out[p*HALF + c];
      }
      sacc[i] += sp; zacc[i] += zp;
    }
  }
  __syncthreads();

  // ---- scores ----
  float nw = 0.f;
  #pragma unroll
  for (int c = 0; c < EMB; ++c) nw += pool_w[c]*pool_w[c];
  const float invn = rsqrtf(nw);
  for (int i = tid; i < Nn; i += BD) {
    skey[i] = tanhf(sacc[i]*invn);
    sval[i] = zacc[i];
  }

  // ---- bitonic sort (ascending) of 1024 (score, z) pairs ----
  for (int k = 2; k <= Nn; k <<= 1)
    for (int j = k >> 1; j > 0; j >>= 1) {
      __syncthreads();
      for (int i = tid; i < Nn; i += BD) {
        int ixj = i ^ j;
        if (ixj > i) {
          float ka = skey[i], kb = skey[ixj];
          bool up = ((i & k) == 0);
          if ((ka > kb) == up) {
            skey[i] = kb; skey[ixj] = ka;
            float va = sval[i]; sval[i] = sval[ixj]; sval[ixj] = va;
          }
        }
      }
    }
  __syncthreads();

  // top-K = sorted tail; relu(out*score).mean @ W_out == (1/K) sum max(s,0)*z
  float sum = 0.f;
  for (int i = tid; i < Nn; i += BD)
    if (i >= Nn - KTOP) sum += fmaxf(skey[i], 0.f) * sval[i];
  red[tid] = sum;
  __syncthreads();
  for (int s = BD/2; s > 0; s >>= 1) {
    if (tid < s) red[tid] += red[tid + s];
    __syncthreads();
  }
  if (tid == 0) out[b] = red[0] / (float)KTOP + b_out[0];
}

extern "C" void kernel_launch(void* const* d_in, const int* in_sizes, int n_in,
                              void* d_out, int out_size, void* d_ws, size_t ws_size,
                              hipStream_t stream) {
  // setup_inputs order: x, edge_index(i64), batch_index(i64, unused),
  //                     W_in, b_in, pool_w, W_out, b_out
  const float* x      = (const float*)d_in[0];
  const int*   e32    = (const int*)  d_in[1];   // int64 viewed as i32 pairs (LE low word)
  const float* W_in   = (const float*)d_in[3];
  const float* b_in   = (const float*)d_in[4];
  const float* pool_w = (const float*)d_in[5];
  const float* W_out  = (const float*)d_in[6];
  const float* b_out  = (const float*)d_in[7];
  float* out = (float*)d_out;

  gcn_topk_fused<<<Bg, BD, 0, stream>>>(x, e32, W_in, b_in, pool_w, W_out, b_out, out);
}